// HeightCompressionMean_8615704395941
// MI455X (gfx1250) — compile-verified
//
#include <hip/hip_runtime.h>

// Height-compression mean for sparse voxels on MI455X (gfx1250, wave32).
//
// out[b,c,h,w] = (1/D) * sum over voxels i with (b_idx,y_idx,x_idx)==(b,h,w)
//                of features[i,c]
//
// Bandwidth-bound scatter: output (128 MiB) is L2-resident (192 MB L2), so
// f32 atomics resolve in L2. Feature reads are streamed with non-temporal
// hints so they don't evict the output working set.

typedef float v4f __attribute__((ext_vector_type(4)));

#define CHANNELS 128  // fixed by the reference (C = 128 = 32 lanes * 4)

__global__ __launch_bounds__(256) void hc_scatter_mean_kernel(
    const float* __restrict__ features,   // [N, 128]
    const int* __restrict__ b_idx,        // [N]
    const int* __restrict__ y_idx,        // [N]
    const int* __restrict__ x_idx,        // [N]
    const int* __restrict__ depth_p,      // [1]
    const int* __restrict__ height_p,     // [1]
    const int* __restrict__ width_p,      // [1]
    float* __restrict__ out,              // [B, 128, H, W]
    int n_voxels)
{
    const int lane          = threadIdx.x & 31;   // wave32
    const int wave_in_block = threadIdx.x >> 5;   // 8 waves per 256-thr block
    const int voxel = blockIdx.x * (blockDim.x >> 5) + wave_in_block;
    if (voxel >= n_voxels) return;                // wave-uniform exit

    // Uniform scalars (broadcast loads from 1-element device arrays).
    const int D = depth_p[0];
    const int H = height_p[0];
    const int W = width_p[0];
    const float inv_d = 1.0f / (float)D;

    // Per-voxel indices: wave-uniform broadcast loads.
    const int b = b_idx[voxel];
    const int y = y_idx[voxel];
    const int x = x_idx[voxel];

    // Lane l loads channels 4l..4l+3: one contiguous 512 B row per wave.
    // Non-temporal: features are read exactly once (204.8 MB stream) --
    // keep them out of the way of the L2-resident output.
    const v4f* src =
        (const v4f*)(features + (size_t)voxel * CHANNELS + (size_t)lane * 4);
    v4f f = __builtin_nontemporal_load(src);

    const size_t HW = (size_t)H * (size_t)W;
    const int c0 = lane * 4;
    // out index: ((b*C + c)*H + y)*W + x
    size_t base =
        (((size_t)b * CHANNELS + (size_t)c0) * (size_t)H + (size_t)y) *
            (size_t)W +
        (size_t)x;
    float* p = out + base;

    // Non-returning native f32 atomics (global_atomic_add_f32, STOREcnt
    // tracked). Channel stride = H*W floats (256 KB) -> distinct L2 lines,
    // but all within the L2-resident output.
    unsafeAtomicAdd(p,          f.x * inv_d);
    unsafeAtomicAdd(p + HW,     f.y * inv_d);
    unsafeAtomicAdd(p + 2 * HW, f.z * inv_d);
    unsafeAtomicAdd(p + 3 * HW, f.w * inv_d);
}

extern "C" void kernel_launch(void* const* d_in, const int* in_sizes, int n_in,
                              void* d_out, int out_size, void* d_ws,
                              size_t ws_size, hipStream_t stream) {
    (void)n_in; (void)d_ws; (void)ws_size;

    const float* features = (const float*)d_in[0];
    const int*   b_idx    = (const int*)d_in[1];
    // d_in[2] = z_idx: mean over depth makes the z position irrelevant to the
    // contribution (each voxel adds features/D regardless of z) -- unused.
    const int*   y_idx    = (const int*)d_in[3];
    const int*   x_idx    = (const int*)d_in[4];
    // d_in[5] = batch_size (implied by out_size; unused on device)
    const int*   depth_p  = (const int*)d_in[6];
    const int*   height_p = (const int*)d_in[7];
    const int*   width_p  = (const int*)d_in[8];

    float* out = (float*)d_out;
    const int n_voxels = in_sizes[1];  // b_idx element count = N

    // Zero-init the accumulation target (graph-capturable memset node).
    hipMemsetAsync(d_out, 0, (size_t)out_size * sizeof(float), stream);

    // One wave32 per voxel; 8 waves (8 voxels) per 256-thread block.
    const int waves_per_block = 8;
    const int blocks = (n_voxels + waves_per_block - 1) / waves_per_block;
    hipLaunchKernelGGL(hc_scatter_mean_kernel, dim3(blocks), dim3(256), 0,
                       stream, features, b_idx, y_idx, x_idx, depth_p,
                       height_p, width_p, out, n_voxels);
}